// GCN_75479755260256
// MI455X (gfx1250) — compile-verified
//
#include <hip/hip_runtime.h>
#include <math.h>

typedef __attribute__((ext_vector_type(2))) float v2f;
typedef __attribute__((ext_vector_type(8))) float v8f;

#define D_FEAT 128
#define XS_STRIDE 132   // padded LDS row stride (floats) to avoid bank conflicts

// ---------------------------------------------------------------- degrees ---
__global__ __launch_bounds__(256) void degree_kernel(
    const int* __restrict__ src, const int* __restrict__ dst,
    float* __restrict__ deg_out, float* __restrict__ deg_in, int nE) {
  int e = blockIdx.x * blockDim.x + threadIdx.x;
  if (e < nE) {
    atomicAdd(&deg_out[src[e]], 1.0f);
    atomicAdd(&deg_in[dst[e]], 1.0f);
  }
}

__global__ __launch_bounds__(256) void isqrt_kernel(float* __restrict__ deg, int n) {
  int i = blockIdx.x * blockDim.x + threadIdx.x;
  if (i < n) {
    float v = deg[i];
    v = v < 1.0f ? 1.0f : v;            // clip(deg, 1.0)
    deg[i] = rsqrtf(v);
  }
}

// ---------------------------------------------------- scaled GEMM via WMMA ---
// Y[n,:] = (X[n,:] * s[n]) @ W   where X: nrows x 128, W: 128 x 128 row-major.
// Block = 256 threads = 8 waves; block computes a 16-row strip, wave w owns
// output columns [16w, 16w+16). K-loop uses V_WMMA_F32_16X16X4_F32 (fp32).
// The 16x128 result tile is staged in LDS and stored coalesced as float4.
__global__ __launch_bounds__(256) void gemm_scaled_wmma(
    const float* __restrict__ X, const float* __restrict__ s,
    const float* __restrict__ W, float* __restrict__ Y, int nrows) {
  __shared__ __align__(16) float xs[16 * XS_STRIDE];  // A tile, reused as D tile

  const int m0   = blockIdx.x * 16;
  const int tid  = threadIdx.x;
  const int wave = tid >> 5;            // 0..7  -> column tile
  const int lane = tid & 31;
  const int half = lane >> 4;           // 0 or 1 (lane group)
  const int l15  = lane & 15;

  // Stage the scaled 16x128 A tile in LDS (one global read per element).
  for (int i = tid; i < 16 * D_FEAT; i += 256) {
    int r = i >> 7, c = i & (D_FEAT - 1);
    int row = m0 + r;
    float v = 0.0f;
    if (row < nrows) v = X[(size_t)row * D_FEAT + c] * s[row];
    xs[r * XS_STRIDE + c] = v;
  }
  __syncthreads();

  const int n0 = wave * 16;
  v8f acc = {};                          // C = 0

  #pragma unroll 4
  for (int k = 0; k < D_FEAT; k += 4) {
    // A fragment (16x4 f32): lanes 0-15 -> K = k+0,k+1 ; lanes 16-31 -> K = k+2,k+3
    int ka = k + 2 * half;
    v2f a, b;
    a.x = xs[l15 * XS_STRIDE + ka];
    a.y = xs[l15 * XS_STRIDE + ka + 1];
    // B fragment (4x16 f32): same K split, N = n0 + l15
    b.x = W[(size_t)(ka)     * D_FEAT + n0 + l15];
    b.y = W[(size_t)(ka + 1) * D_FEAT + n0 + l15];
    acc = __builtin_amdgcn_wmma_f32_16x16x4_f32(
        /*neg_a=*/false, a, /*neg_b=*/false, b,
        /*c_mod=*/(short)0, acc, /*reuse_a=*/false, /*reuse_b=*/false);
  }

  __syncthreads();   // all A-tile reads done; reuse xs as the D tile

  // D layout: VGPR v -> M = v + 8*half, N = l15. Lanes in a half-wave write
  // consecutive columns -> conflict-free LDS stores.
  #pragma unroll
  for (int v = 0; v < 8; ++v)
    xs[(v + 8 * half) * D_FEAT + n0 + l15] = acc[v];
  __syncthreads();

  // Coalesced store: the block tile is a contiguous 16*128-float span of Y.
  if (m0 + 16 <= nrows) {
    float4* __restrict__ Y4 = (float4*)(Y + (size_t)m0 * D_FEAT);
    const float4* t4 = (const float4*)xs;
    #pragma unroll
    for (int i = tid; i < (16 * D_FEAT) / 4; i += 256)
      Y4[i] = t4[i];
  } else {
    for (int i = tid; i < 16 * D_FEAT; i += 256) {
      int r = i >> 7;
      if (m0 + r < nrows) Y[(size_t)m0 * D_FEAT + i] = xs[i];
    }
  }
}

// -------------------------------------------------------------------- SpMM ---
// agg[dst[e], :] += t[src[e], :] ; one wave per edge, float4 per lane.
__global__ __launch_bounds__(256) void spmm_kernel(
    const int* __restrict__ src, const int* __restrict__ dst,
    const float* __restrict__ t, float* __restrict__ agg, int nE) {
  int gid  = blockIdx.x * blockDim.x + threadIdx.x;
  int e    = gid >> 5;
  if (e >= nE) return;
  int lane = gid & 31;
  int sn = src[e];
  int dn = dst[e];
  const float4 v = ((const float4*)(t + (size_t)sn * D_FEAT))[lane];
  float* out = agg + (size_t)dn * D_FEAT + lane * 4;
  atomicAdd(out + 0, v.x);
  atomicAdd(out + 1, v.y);
  atomicAdd(out + 2, v.z);
  atomicAdd(out + 3, v.w);
}

// ---------------------------------------------------- scale + bias + ReLU ---
__global__ __launch_bounds__(256) void post_kernel(
    float* __restrict__ h, const float* __restrict__ in_isqrt,
    const float* __restrict__ bias, int nN) {
  int i = blockIdx.x * blockDim.x + threadIdx.x;
  if (i >= nN * D_FEAT) return;
  int n = i >> 7, f = i & (D_FEAT - 1);
  float v = h[i] * in_isqrt[n] + bias[f];
  h[i] = v > 0.0f ? v : 0.0f;
}

// ----------------------------------------------------------------- pooling ---
__global__ __launch_bounds__(256) void counts_kernel(
    const int* __restrict__ n2g, float* __restrict__ counts, int nN) {
  int i = blockIdx.x * blockDim.x + threadIdx.x;
  if (i < nN) atomicAdd(&counts[n2g[i]], 1.0f);
}

__global__ __launch_bounds__(256) void pool_kernel(
    const float* __restrict__ h, const int* __restrict__ n2g,
    float* __restrict__ sums, int nN, int G) {
  __shared__ float acc[16 * D_FEAT];
  int lim = (G <= 16 ? G : 16) * D_FEAT;
  for (int i = threadIdx.x; i < lim; i += blockDim.x) acc[i] = 0.0f;
  __syncthreads();

  int f   = threadIdx.x & (D_FEAT - 1);
  int sub = threadIdx.x >> 7;          // 0..1
  int base = blockIdx.x * 128;
  for (int r = sub; r < 128; r += 2) {
    int n = base + r;
    if (n < nN) {
      int g = n2g[n];
      float v = h[(size_t)n * D_FEAT + f];
      if (g < 16) atomicAdd(&acc[g * D_FEAT + f], v);     // ds_add_f32
      else        atomicAdd(&sums[(size_t)g * D_FEAT + f], v);
    }
  }
  __syncthreads();
  for (int i = threadIdx.x; i < lim; i += blockDim.x) {
    float v = acc[i];
    if (v != 0.0f) atomicAdd(&sums[i], v);
  }
}

// ------------------------------------------------------ head + softmax ------
__global__ void head_kernel(const float* __restrict__ sums,
                            const float* __restrict__ counts,
                            const float* __restrict__ Wl,
                            const float* __restrict__ bl,
                            float* __restrict__ out, int G, int C) {
  int g = threadIdx.x;
  if (g >= G) return;
  float cnt = counts[g];
  cnt = cnt < 1.0f ? 1.0f : cnt;
  float inv = 1.0f / cnt;

  float logits[32];
  float mx = -3.0e38f;
  for (int c = 0; c < C && c < 32; ++c) {
    float a = bl[c];
    for (int d = 0; d < D_FEAT; ++d)
      a += sums[(size_t)g * D_FEAT + d] * inv * Wl[(size_t)d * C + c];
    logits[c] = a;
    mx = fmaxf(mx, a);
  }
  float se = 0.0f;
  for (int c = 0; c < C && c < 32; ++c) {
    float e = expf(logits[c] - mx);
    logits[c] = e;
    se += e;
  }
  float rs = 1.0f / se;
  for (int c = 0; c < C && c < 32; ++c)
    out[(size_t)g * C + c] = logits[c] * rs;
}

// ---------------------------------------------------------------- launcher ---
extern "C" void kernel_launch(void* const* d_in, const int* in_sizes, int n_in,
                              void* d_out, int out_size, void* d_ws, size_t ws_size,
                              hipStream_t stream) {
  (void)n_in; (void)ws_size;
  const float* x   = (const float*)d_in[0];
  const int*   ei  = (const int*)  d_in[1];
  const int*   n2g = (const int*)  d_in[2];
  const float* W1  = (const float*)d_in[3];
  const float* b1  = (const float*)d_in[4];
  const float* W2  = (const float*)d_in[5];
  const float* b2  = (const float*)d_in[6];
  const float* Wl  = (const float*)d_in[7];
  const float* bl  = (const float*)d_in[8];

  const int N = in_sizes[0] / D_FEAT;
  const int E = in_sizes[1] / 2;
  const int C = in_sizes[8];
  const int G = out_size / C;

  const int* src = ei;           // edge_index row 0
  const int* dst = ei + E;       // edge_index row 1

  // workspace layout (floats): [deg_out|deg_in][bufA][bufB][sums][counts]
  float* ws       = (float*)d_ws;
  float* deg      = ws;                                    // 2*N
  float* bufA     = ws + 2 * (size_t)N;                    // N*128
  float* bufB     = bufA + (size_t)N * D_FEAT;             // N*128
  float* sums     = bufB + (size_t)N * D_FEAT;             // G*128
  float* counts   = sums + (size_t)G * D_FEAT;             // G
  float* out_isqrt = deg;
  float* in_isqrt  = deg + N;

  const int T = 256;
  // --- degree normalization ---
  hipMemsetAsync(deg, 0, sizeof(float) * 2 * (size_t)N, stream);
  degree_kernel<<<(E + T - 1) / T, T, 0, stream>>>(src, dst, out_isqrt, in_isqrt, E);
  isqrt_kernel<<<(2 * N + T - 1) / T, T, 0, stream>>>(deg, 2 * N);

  // --- layer 1: relu( SpMM(A, (x*so)@W1) * si + b1 ) ---
  gemm_scaled_wmma<<<(N + 15) / 16, T, 0, stream>>>(x, out_isqrt, W1, bufA, N);
  hipMemsetAsync(bufB, 0, sizeof(float) * (size_t)N * D_FEAT, stream);
  spmm_kernel<<<((size_t)E * 32 + T - 1) / T, T, 0, stream>>>(src, dst, bufA, bufB, E);
  post_kernel<<<((size_t)N * D_FEAT + T - 1) / T, T, 0, stream>>>(bufB, in_isqrt, b1, N);

  // --- layer 2 ---
  gemm_scaled_wmma<<<(N + 15) / 16, T, 0, stream>>>(bufB, out_isqrt, W2, bufA, N);
  hipMemsetAsync(bufB, 0, sizeof(float) * (size_t)N * D_FEAT, stream);
  spmm_kernel<<<((size_t)E * 32 + T - 1) / T, T, 0, stream>>>(src, dst, bufA, bufB, E);
  post_kernel<<<((size_t)N * D_FEAT + T - 1) / T, T, 0, stream>>>(bufB, in_isqrt, b2, N);

  // --- average pooling + linear head + softmax ---
  hipMemsetAsync(sums, 0, sizeof(float) * ((size_t)G * D_FEAT + G), stream);
  counts_kernel<<<(N + T - 1) / T, T, 0, stream>>>(n2g, counts, N);
  pool_kernel<<<(N + 127) / 128, T, 0, stream>>>(bufB, n2g, sums, N, G);
  head_kernel<<<1, 32, 0, stream>>>(sums, counts, Wl, bl, (float*)d_out, G, C);
}